// SeqEmbCNN_81750407512559
// MI455X (gfx1250) — compile-verified
//
#include <hip/hip_runtime.h>
#include <math.h>

typedef __attribute__((ext_vector_type(2))) float v2f;
typedef __attribute__((ext_vector_type(8))) float v8f;

#define B_      4096
#define L_      512
#define DEMB    8
#define NF      32
#define DOUT    64
#define TOTCODES 3900   // 25 + 125 + 625 + 3125

__device__ __forceinline__ float gelu_exact(float x) {
    return 0.5f * x * (1.0f + erff(x * 0.70710678118654752f));
}

// ---------------------------------------------------------------------------
// Kernel A: build per-branch LUTs.  One wave == one window-code; lane == filter.
//   h_tab[code][f] = gelu(conv(code)[f] + b[f])
//   s_tab[code]    = dot(h_tab[code], att_w) + att_b
// ---------------------------------------------------------------------------
__global__ __launch_bounds__(256)
void lut_kernel(const float* __restrict__ emb,
                const float* __restrict__ w0, const float* __restrict__ b0,
                const float* __restrict__ aw0, const float* __restrict__ ab0,
                const float* __restrict__ w1, const float* __restrict__ b1,
                const float* __restrict__ aw1, const float* __restrict__ ab1,
                const float* __restrict__ w2, const float* __restrict__ b2,
                const float* __restrict__ aw2, const float* __restrict__ ab2,
                const float* __restrict__ w3, const float* __restrict__ b3,
                const float* __restrict__ aw3, const float* __restrict__ ab3,
                float* __restrict__ htab, float* __restrict__ stab)
{
    int gid = blockIdx.x * blockDim.x + threadIdx.x;
    int cg  = gid >> 5;          // global code index (uniform across wave)
    int f   = gid & 31;          // filter (lane)
    if (cg >= TOTCODES) return;

    int br, code;
    const float *W, *Bv, *AW, *AB;
    if      (cg < 25)  { br = 0; code = cg;       W = w0; Bv = b0; AW = aw0; AB = ab0; }
    else if (cg < 150) { br = 1; code = cg - 25;  W = w1; Bv = b1; AW = aw1; AB = ab1; }
    else if (cg < 775) { br = 2; code = cg - 150; W = w2; Bv = b2; AW = aw2; AB = ab2; }
    else               { br = 3; code = cg - 775; W = w3; Bv = b3; AW = aw3; AB = ab3; }
    int ks = br + 2;

    int d[5]; int rem = code;
    for (int t = 0; t < ks; ++t) { d[t] = rem % 5; rem /= 5; }

    float s = Bv[f];
    for (int t = 0; t < ks; ++t) {
        const float* erow = emb + d[t] * DEMB;   // pad row is already zero
        for (int c = 0; c < DEMB; ++c)
            s = fmaf(W[(f * DEMB + c) * ks + t], erow[c], s);
    }
    float g = gelu_exact(s);
    htab[cg * NF + f] = g;

    float p = g * AW[f];
    for (int off = 16; off > 0; off >>= 1) p += __shfl_xor(p, off, 32);
    if (f == 0) stab[cg] = p + AB[0];
}

// ---------------------------------------------------------------------------
// Kernel B: per batch-row softmax-attention pooling via LUT gathers.
// One block per b; 8 waves; waves (2i, 2i+1) handle branch i.
// ---------------------------------------------------------------------------
__global__ __launch_bounds__(256)
void pool_kernel(const int* __restrict__ xidx,
                 const float* __restrict__ stab,
                 const float* __restrict__ htab,
                 float* __restrict__ zout)
{
    __shared__ int            xrow[L_];
    __shared__ float          stab_s[TOTCODES];
    __shared__ float          sval[4][L_];
    __shared__ unsigned short cod[4][L_];
    __shared__ float          red_m[4][2];
    __shared__ float          red_s[4][2];
    __shared__ float          acc2[4][NF];

    const int tid = threadIdx.x;
    const int b   = blockIdx.x;
    const int* xr = xidx + (long)b * L_;

    for (int i = tid; i < L_;       i += 256) xrow[i]   = xr[i];
    for (int i = tid; i < TOTCODES; i += 256) stab_s[i] = stab[i];
    __syncthreads();

    const int wave = tid >> 5, lane = tid & 31;
    const int br = wave >> 1, wp = wave & 1;
    const int ks = br + 2;
    const int Lp = L_ - ks + 1;
    const int soff = (br == 0) ? 0 : (br == 1) ? 25 : (br == 2) ? 150 : 775;
    const int tp = wp * 32 + lane;

    // pass 1: codes, scores, running max
    float m = -3.4e38f;
    for (int l = tp; l < Lp; l += 64) {
        int code = 0, pw = 1;
        for (int t = 0; t < ks; ++t) { code += xrow[l + t] * pw; pw *= 5; }
        float s = stab_s[soff + code];
        sval[br][l] = s;
        cod[br][l]  = (unsigned short)code;
        m = fmaxf(m, s);
    }
    for (int off = 16; off > 0; off >>= 1) m = fmaxf(m, __shfl_xor(m, off, 32));
    if (lane == 0) red_m[br][wp] = m;
    __syncthreads();
    m = fmaxf(red_m[br][0], red_m[br][1]);

    // pass 2: exp + sum
    float zs = 0.f;
    for (int l = tp; l < Lp; l += 64) {
        float e = __expf(sval[br][l] - m);
        sval[br][l] = e;
        zs += e;
    }
    for (int off = 16; off > 0; off >>= 1) zs += __shfl_xor(zs, off, 32);
    if (lane == 0) red_s[br][wp] = zs;
    __syncthreads();
    const float Z = red_s[br][0] + red_s[br][1];

    // pass 3: weighted pooling; lane == filter, h_tab row read is a 128B burst
    const float* ht = htab + (long)soff * NF;
    float a = 0.f;
    for (int l = wp; l < Lp; l += 2) {
        float w   = sval[br][l];            // LDS broadcast
        int  code = cod[br][l];
        if (l + 8 < Lp)
            __builtin_prefetch(ht + (int)cod[br][l + 8] * NF, 0, 0);
        a = fmaf(w, ht[code * NF + lane], a);
    }
    if (wp == 1) acc2[br][lane] = a;
    __syncthreads();
    if (wp == 0) {
        a = (a + acc2[br][lane]) / Z;
        zout[(long)b * 128 + br * NF + lane] = a;
    }
}

// ---------------------------------------------------------------------------
// Kernel C: z[4096x128] @ proj_w^T[128x64] + bias -> GELU -> LayerNorm.
// fp32 WMMA 16x16x4. One wave owns a 16(M) x 64(N) stripe: 4 C tiles, K=128.
// ---------------------------------------------------------------------------
__global__ __launch_bounds__(128)
void proj_kernel(const float* __restrict__ z,
                 const float* __restrict__ pw,
                 const float* __restrict__ pb,
                 const float* __restrict__ gamma,
                 const float* __restrict__ beta,
                 float* __restrict__ out)
{
    __shared__ float pws[DOUT * 130];   // +2 pad: kills stride-128 bank conflicts

    const int tid = threadIdx.x;
    for (int i = tid; i < DOUT * 128; i += 128)
        pws[(i >> 7) * 130 + (i & 127)] = pw[i];
    __syncthreads();

    const int wave  = tid >> 5, lane = tid & 31;
    const int mtile = blockIdx.x * 4 + wave;       // 64 blocks x 4 waves = 256 tiles
    const int row   = lane & 15;                   // A-matrix M row for this lane
    const int col   = lane & 15;                   // B/C N column for this lane
    const int khalf = (lane >> 4) << 1;            // lanes 16-31 hold K+2,K+3

    const float* arow = z + (long)(mtile * 16 + row) * 128;

    v8f acc[4] = {};
    for (int k0 = 0; k0 < 128; k0 += 4) {
        const int k = k0 + khalf;
        v2f a; a[0] = arow[k]; a[1] = arow[k + 1];
#pragma unroll
        for (int n = 0; n < 4; ++n) {
            v2f bb;
            const int cb = (n * 16 + col) * 130 + k;  // B[k][N] = proj_w[N][k]
            bb[0] = pws[cb]; bb[1] = pws[cb + 1];
            acc[n] = __builtin_amdgcn_wmma_f32_16x16x4_f32(
                false, a, false, bb, (short)0, acc[n], false, false);
        }
    }

    // epilogue: bias + exact GELU + LayerNorm over the 64 columns of each row
    float pbv[4], gv[4], bv[4];
#pragma unroll
    for (int n = 0; n < 4; ++n) {
        pbv[n] = pb[n * 16 + col];
        gv[n]  = gamma[n * 16 + col];
        bv[n]  = beta[n * 16 + col];
    }
    float vals[4][8];
#pragma unroll
    for (int n = 0; n < 4; ++n)
#pragma unroll
        for (int r = 0; r < 8; ++r)
            vals[n][r] = gelu_exact(acc[n][r] + pbv[n]);

    const int mbase = mtile * 16 + ((lane >> 4) ? 8 : 0);
#pragma unroll
    for (int r = 0; r < 8; ++r) {
        float t = 0.f, t2 = 0.f;
#pragma unroll
        for (int n = 0; n < 4; ++n) { t += vals[n][r]; t2 += vals[n][r] * vals[n][r]; }
        // reduce across the 16 lanes holding this row (half-wave local)
        for (int off = 8; off > 0; off >>= 1) {
            t  += __shfl_xor(t,  off, 16);
            t2 += __shfl_xor(t2, off, 16);
        }
        const float mu  = t * (1.0f / 64.0f);
        const float var = t2 * (1.0f / 64.0f) - mu * mu;
        const float inv = rsqrtf(var + 1e-5f);
#pragma unroll
        for (int n = 0; n < 4; ++n) {
            float v = (vals[n][r] - mu) * inv * gv[n] + bv[n];
            out[(long)(mbase + r) * 64 + n * 16 + col] = v;
        }
    }
}

// ---------------------------------------------------------------------------
extern "C" void kernel_launch(void* const* d_in, const int* in_sizes, int n_in,
                              void* d_out, int out_size, void* d_ws, size_t ws_size,
                              hipStream_t stream)
{
    const int*   x_idx = (const int*)  d_in[0];
    const float* emb   = (const float*)d_in[1];
    const float* cw[4], *cb[4], *aw[4], *ab[4];
    for (int i = 0; i < 4; ++i) {
        cw[i] = (const float*)d_in[2 + 4 * i];
        cb[i] = (const float*)d_in[3 + 4 * i];
        aw[i] = (const float*)d_in[4 + 4 * i];
        ab[i] = (const float*)d_in[5 + 4 * i];
    }
    const float* proj_w = (const float*)d_in[18];
    const float* proj_b = (const float*)d_in[19];
    const float* gamma  = (const float*)d_in[20];
    const float* beta   = (const float*)d_in[21];
    float* out = (float*)d_out;

    float* wsf  = (float*)d_ws;
    float* htab = wsf;                       // 3900*32 = 124800 floats
    float* stab = wsf + 124800;              // 3900 floats
    float* zbuf = wsf + 128700;              // 4096*128 = 524288 floats

    lut_kernel<<<(TOTCODES * 32 + 255) / 256, 256, 0, stream>>>(
        emb,
        cw[0], cb[0], aw[0], ab[0],
        cw[1], cb[1], aw[1], ab[1],
        cw[2], cb[2], aw[2], ab[2],
        cw[3], cb[3], aw[3], ab[3],
        htab, stab);

    pool_kernel<<<B_, 256, 0, stream>>>(x_idx, stab, htab, zbuf);

    proj_kernel<<<64, 128, 0, stream>>>(zbuf, proj_w, proj_b, gamma, beta, out);
}